// AdaptiveLocalPooling_25039659336081
// MI455X (gfx1250) — compile-verified
//
#include <hip/hip_runtime.h>
#include <math.h>

// AdaptiveLocalPooling on MI455X (gfx1250)
// B=16, T=4096, C=384, K=9.  Gather-bandwidth bound; X (100MB) is L2-resident.
// Strategy: wave32 per t.  The 9 indexed neighbor rows are fetched with ONE
// Tensor Data Mover gather (tensor_load_to_lds, gather_mode, 16-bit indices),
// double-buffered in LDS and pipelined under s_wait_tensorcnt so the DMA for
// tile t+1 overlaps all compute of tile t.  Deterministic two-kernel mean.

#define CC     384
#define KN     9
#define WAVES  2
#define TPB    (WAVES * 32)
#define BATCH  16
#define TDIM   4096
#define EPSV   1e-8f

typedef unsigned int u32x4 __attribute__((ext_vector_type(4)));
typedef int          i32x4 __attribute__((ext_vector_type(4)));
typedef int          i32x8 __attribute__((ext_vector_type(8)));

__device__ __forceinline__ float wsum32(float v) {
#pragma unroll
  for (int m = 16; m >= 1; m >>= 1) v += __shfl_xor(v, m, 32);
  return v;
}

__device__ __forceinline__ float dot4f(float4 a, float4 b) {
  return a.x * b.x + a.y * b.y + a.z * b.z + a.w * b.w;
}

// Issue the TDM op via inline asm (portable across both toolchains' builtin
// arities).  Operands: D# group0 (4 SGPRs), group1 (8 SGPRs), group2 (4),
// group3 (4) -- groups 2/3 carry the packed 16-bit gather row indices.
__device__ __forceinline__ void tdm_issue(u32x4 g0, i32x8 g1, i32x4 g2, i32x4 g3) {
  asm volatile("tensor_load_to_lds %0, %1, %2, %3"
               :: "s"(g0), "s"(g1), "s"(g2), "s"(g3)
               : "memory");
}

// One TDM gather: 9 rows of 384 f32, row indices packed as 16-bit, into LDS.
__device__ __forceinline__ void issue_gather(uint32_t lds_addr,
                                             uint32_t ga_lo, uint32_t ga_hi,
                                             i32x8 g1,
                                             const int* __restrict__ ip) {
  const int i0 = ip[0], i1 = ip[1], i2 = ip[2], i3 = ip[3], i4 = ip[4];
  const int i5 = ip[5], i6 = ip[6], i7 = ip[7], i8 = ip[8];
  i32x4 g2;
  g2[0] = __builtin_amdgcn_readfirstlane(i0 | (i1 << 16));
  g2[1] = __builtin_amdgcn_readfirstlane(i2 | (i3 << 16));
  g2[2] = __builtin_amdgcn_readfirstlane(i4 | (i5 << 16));
  g2[3] = __builtin_amdgcn_readfirstlane(i6 | (i7 << 16));
  i32x4 g3;
  g3[0] = __builtin_amdgcn_readfirstlane(i8);
  g3[1] = 0; g3[2] = 0; g3[3] = 0;
  u32x4 g0;
  g0[0] = 0x80000001u;      // count=1, gather_mode=1, 16-bit indices
  g0[1] = lds_addr;         // LDS byte address of tile
  g0[2] = ga_lo;            // tensor base addr [31:0]
  g0[3] = ga_hi;            // addr[56:32] | type=2 in [127:126]
  tdm_issue(g0, g1, g2, g3);
}

__global__ __launch_bounds__(TPB)
void alp_pool_kernel(const float* __restrict__ X,
                     const int* __restrict__ nidx,
                     float* __restrict__ part,
                     int tpw, int nchunks) {
  __shared__ float smem[WAVES][2][KN * CC];   // 2*2*3456*4B = 55296 B

  const int lane  = (int)(threadIdx.x & 31u);
  const int wave  = (int)(threadIdx.x >> 5u);
  const int b     = (int)blockIdx.y;
  const int chunk = (int)blockIdx.x;

  const float* Xb = X + (size_t)b * TDIM * CC;
  const uint64_t xbase = (uint64_t)(uintptr_t)Xb;
  const uint32_t ga_lo = (uint32_t)xbase;
  const uint32_t ga_hi = ((uint32_t)(xbase >> 32) & 0x01FFFFFFu) | 0x80000000u;

  // low 32 bits of flat shared address == LDS byte offset
  uint32_t lds_lo = (uint32_t)(uintptr_t)&smem[wave][0][0];
  uint32_t lds_hi = (uint32_t)(uintptr_t)&smem[wave][1][0];
  lds_lo = (uint32_t)__builtin_amdgcn_readfirstlane((int)lds_lo);
  lds_hi = (uint32_t)__builtin_amdgcn_readfirstlane((int)lds_hi);

  // D# group1: data_size=4B, tensor_dim0=384, tensor_dim1=4096,
  // tile_dim0=384, tile_dim1=9 rows, dim0_stride=384 elements.
  i32x8 g1;
  g1[0] = 0x00020000;          // data_size = 2 (4 bytes)
  g1[1] = (int)(CC << 16);     // tensor_dim0[15:0] << 16
  g1[2] = (int)(TDIM << 16);   // tensor_dim1[15:0] << 16
  g1[3] = (int)(CC << 16);     // tile_dim0 << 16
  g1[4] = KN;                  // tile_dim1 = #valid gather indices
  g1[5] = CC;                  // tensor_dim0_stride
  g1[6] = 0;
  g1[7] = 0;

  float4 acc0 = make_float4(0.f, 0.f, 0.f, 0.f);
  float4 acc1 = make_float4(0.f, 0.f, 0.f, 0.f);
  float4 acc2 = make_float4(0.f, 0.f, 0.f, 0.f);

  const int tbase = chunk * (WAVES * tpw) + wave * tpw;

  // prologue: DMA the first tile's 9 rows into buffer 0
  issue_gather(lds_lo, ga_lo, ga_hi, g1, nidx + (size_t)tbase * KN);

  for (int tt = 0; tt < tpw; ++tt) {
    const int t = tbase + tt;

    // prefetch next tile into the other buffer (DMA overlaps this tile's work)
    if (tt + 1 < tpw) {
      asm volatile("s_wait_dscnt 0x0" ::: "memory");  // prior reads retired
      issue_gather((tt & 1) ? lds_lo : lds_hi, ga_lo, ga_hi, g1,
                   nidx + (size_t)(t + 1) * KN);
    }

    // query row straight to VGPRs; ||q|| via wave butterfly
    const float4* xr = (const float4*)(Xb + (size_t)t * CC);
    const float4 x0 = xr[lane];
    const float4 x1 = xr[lane + 32];
    const float4 x2 = xr[lane + 64];
    const float q2 = dot4f(x0, x0) + dot4f(x1, x1) + dot4f(x2, x2);
    const float qn = fmaxf(sqrtf(wsum32(q2)), EPSV);

    // wait for this tile's gather (in-order: <=1 leaves the prefetch in flight)
    if (tt + 1 < tpw) __builtin_amdgcn_s_wait_tensorcnt(1);
    else              __builtin_amdgcn_s_wait_tensorcnt(0);
    asm volatile("" ::: "memory");

    const float* stage = &smem[wave][tt & 1][0];

    // dots + neighbor norms
    float sim[KN];
#pragma unroll
    for (int k = 0; k < KN; ++k) {
      const float4* nr = (const float4*)(stage + k * CC);
      const float4 a0 = nr[lane];
      const float4 a1 = nr[lane + 32];
      const float4 a2 = nr[lane + 64];
      const float dp = dot4f(x0, a0) + dot4f(x1, a1) + dot4f(x2, a2);
      const float np = dot4f(a0, a0) + dot4f(a1, a1) + dot4f(a2, a2);
      const float d  = wsum32(dp);
      const float n2 = wsum32(np);
      sim[k] = d / (qn * fmaxf(sqrtf(n2), EPSV));
    }

    // softmax over K (replicated in every lane)
    float m = sim[0];
#pragma unroll
    for (int k = 1; k < KN; ++k) m = fmaxf(m, sim[k]);
    float w[KN];
    float s = 0.f;
#pragma unroll
    for (int k = 0; k < KN; ++k) { w[k] = __expf(sim[k] - m); s += w[k]; }
    const float inv = 1.f / s;

    // weighted pooling
#pragma unroll
    for (int k = 0; k < KN; ++k) {
      const float wk = w[k] * inv;
      const float4* nr = (const float4*)(stage + k * CC);
      const float4 a0 = nr[lane];
      const float4 a1 = nr[lane + 32];
      const float4 a2 = nr[lane + 64];
      acc0.x += wk * a0.x; acc0.y += wk * a0.y;
      acc0.z += wk * a0.z; acc0.w += wk * a0.w;
      acc1.x += wk * a1.x; acc1.y += wk * a1.y;
      acc1.z += wk * a1.z; acc1.w += wk * a1.w;
      acc2.x += wk * a2.x; acc2.y += wk * a2.y;
      acc2.z += wk * a2.z; acc2.w += wk * a2.w;
    }
  }

  // combine waves in fixed order; write per-(b,chunk) partial
  __syncthreads();
  float* red = &smem[0][0][0];               // reuse staging LDS (WAVES*CC f32)
  float4* rw = (float4*)(red + wave * CC);
  rw[lane]      = acc0;
  rw[lane + 32] = acc1;
  rw[lane + 64] = acc2;
  __syncthreads();
#pragma unroll
  for (int i = 0; i < (CC / TPB); ++i) {
    const int c = i * TPB + (int)threadIdx.x;
    float v = 0.f;
#pragma unroll
    for (int wv = 0; wv < WAVES; ++wv) v += red[wv * CC + c];
    part[((size_t)b * nchunks + chunk) * CC + c] = v;
  }
}

__global__ __launch_bounds__(CC)
void alp_reduce_kernel(const float* __restrict__ part,
                       float* __restrict__ out, int nchunks) {
  const int b = (int)blockIdx.x;
  const int c = (int)threadIdx.x;
  float s = 0.f;
  for (int ch = 0; ch < nchunks; ++ch)
    s += part[((size_t)b * nchunks + ch) * CC + c];
  out[b * CC + c] = s * (1.0f / (float)TDIM);
}

extern "C" void kernel_launch(void* const* d_in, const int* in_sizes, int n_in,
                              void* d_out, int out_size, void* d_ws, size_t ws_size,
                              hipStream_t stream) {
  (void)in_sizes; (void)n_in; (void)out_size;
  const float* X    = (const float*)d_in[0];
  const int*   nidx = (const int*)d_in[1];
  float*       out  = (float*)d_out;
  float*       part = (float*)d_ws;

  // pick chunk count so partials fit in workspace (B * nchunks * CC floats)
  int nchunks = 64;
  while (nchunks > 1 &&
         (size_t)BATCH * (size_t)nchunks * CC * sizeof(float) > ws_size)
    nchunks >>= 1;
  const int tpw = TDIM / (nchunks * WAVES);   // t's per wave

  dim3 grid((unsigned)nchunks, BATCH);
  alp_pool_kernel<<<grid, TPB, 0, stream>>>(X, nidx, part, tpw, nchunks);
  alp_reduce_kernel<<<BATCH, CC, 0, stream>>>(part, out, nchunks);
}